// GCNNodeClassification_4861902979273
// MI455X (gfx1250) — compile-verified
//
#include <hip/hip_runtime.h>
#include <hip/hip_bf16.h>

typedef __attribute__((ext_vector_type(2))) float v2f;
typedef __attribute__((ext_vector_type(8))) float v8f;

#define HID 128
#define LSTRIDE (HID + 2)   // padded LDS stride (floats) to avoid bank conflicts

// ---------------- degree / norm kernels ----------------
__global__ void k_fill1(float* __restrict__ deg, int n) {
  int i = blockIdx.x * blockDim.x + threadIdx.x;
  if (i < n) deg[i] = 1.0f;                    // self-loop contributes 1
}

__global__ void k_deg(const int* __restrict__ dst, float* __restrict__ deg, int e) {
  int i = blockIdx.x * blockDim.x + threadIdx.x;
  if (i < e) atomicAdd(&deg[dst[i]], 1.0f);
}

__global__ void k_rsqrt(float* __restrict__ d, int n) {
  int i = blockIdx.x * blockDim.x + threadIdx.x;
  if (i < n) d[i] = rsqrtf(d[i]);              // deg >= 1 always (self-loops)
}

// ---------------- WMMA GEMM: [nrows x 128] @ [128 x 128] ----------------
// W is transposed into LDS (sWT[n][k], stride 130) so a B fragment
// {W[kb][n], W[kb+1][n]} is one aligned ds_load_b64 -> even VGPR pair.
// A-frag (16x4 f32): lane l -> row m=l&15, K-pair 2*(l>>4). B-frag mirrors.
// D (16x16 f32): lane l, vgpr r -> D[8*(l>>4)+r][l&15].
__global__ __launch_bounds__(256) void k_gemm128(const float* __restrict__ A,
                                                 const float* __restrict__ W,
                                                 float* __restrict__ Out, int nrows) {
  __shared__ __align__(16) float sWT[HID * LSTRIDE];   // ~65 KB
  const int t = threadIdx.x;
  {
    const float4* Wv = (const float4*)W;               // W row-major [k][n]
    for (int i = t; i < HID * HID / 4; i += 256) {
      float4 w = Wv[i];
      int k = i >> 5;                                  // 32 float4 per row
      int n0 = (i & 31) * 4;
      sWT[(n0 + 0) * LSTRIDE + k] = w.x;
      sWT[(n0 + 1) * LSTRIDE + k] = w.y;
      sWT[(n0 + 2) * LSTRIDE + k] = w.z;
      sWT[(n0 + 3) * LSTRIDE + k] = w.w;
    }
  }
  __syncthreads();

  const int wave = t >> 5, lane = t & 31;
  const int row0 = (blockIdx.x * 8 + wave) * 16;
  if (row0 >= nrows) return;                   // wave-uniform; EXEC all-1s at WMMA

  const int m = lane & 15;
  const int khalf = lane >> 4;                 // 0: K={0,1}, 1: K={2,3}
  v8f acc[8] = {};
  const float* Arow = A + (size_t)(row0 + m) * HID;

  for (int k0 = 0; k0 < HID; k0 += 4) {
    const int kb = k0 + 2 * khalf;
    v2f a = *(const v2f*)(Arow + kb);          // global_load_b64
#pragma unroll
    for (int nt = 0; nt < 8; ++nt) {
      v2f b = *(const v2f*)(sWT + (nt * 16 + m) * LSTRIDE + kb);  // ds_load_b64
      acc[nt] = __builtin_amdgcn_wmma_f32_16x16x4_f32(
          false, a, false, b, (short)0, acc[nt], false, false);
    }
  }

#pragma unroll
  for (int nt = 0; nt < 8; ++nt) {
    const int n = nt * 16 + m;
#pragma unroll
    for (int r = 0; r < 8; ++r)
      Out[(size_t)(row0 + khalf * 8 + r) * HID + n] = acc[nt][r];
  }
}

// ---------------- agg init: fold self-loop, agg[i] = t[i] * dinv[i]^2 ----------------
__global__ void k_agg_init(const float* __restrict__ t, const float* __restrict__ dinv,
                           float* __restrict__ agg, int n) {
  int idx = blockIdx.x * blockDim.x + threadIdx.x;   // one float4 per thread
  int node = idx >> 5, q = idx & 31;
  if (node >= n) return;
  float w = dinv[node];
  w = w * w;
  float4 v = ((const float4*)(t + (size_t)node * HID))[q];
  v.x *= w; v.y *= w; v.z *= w; v.w *= w;
  ((float4*)(agg + (size_t)node * HID))[q] = v;
}

// ---------------- edge scatter: one wave per edge (128 floats = 32 lanes x float4) ----------------
__global__ void k_edge(const int* __restrict__ src, const int* __restrict__ dst,
                       const float* __restrict__ dinv, const float* __restrict__ t,
                       float* __restrict__ agg, int e) {
  int gid = blockIdx.x * blockDim.x + threadIdx.x;
  int edge = gid >> 5, lane = gid & 31;
  if (edge >= e) return;
  int s = src[edge], d = dst[edge];
  float norm = dinv[s] * dinv[d];
  float4 v = ((const float4*)(t + (size_t)s * HID))[lane];  // L2-resident gather
  float* ad = agg + (size_t)d * HID + lane * 4;
  atomicAdd(ad + 0, v.x * norm);
  atomicAdd(ad + 1, v.y * norm);
  atomicAdd(ad + 2, v.z * norm);
  atomicAdd(ad + 3, v.w * norm);
}

// ---------------- bias + relu, in place ----------------
__global__ void k_bias_relu(float* __restrict__ h, const float* __restrict__ bias,
                            unsigned long long total) {
  unsigned long long i = (unsigned long long)blockIdx.x * blockDim.x + threadIdx.x;
  if (i >= total) return;
  float v = h[i] + bias[(int)(i & (HID - 1))];
  h[i] = v > 0.0f ? v : 0.0f;
}

// ---------------- classifier: [nrows x 128] @ [128 x 40] + bh ----------------
__global__ __launch_bounds__(256) void k_cls(const float* __restrict__ H,
                                             const float* __restrict__ Wh,
                                             const float* __restrict__ bh,
                                             float* __restrict__ Out, int nrows) {
  __shared__ __align__(16) float sWT[48 * LSTRIDE];    // 40 cols padded to 48, transposed
  const int t = threadIdx.x;
  for (int i = t; i < HID * 48; i += 256) {
    int k = i / 48, nn = i % 48;
    sWT[nn * LSTRIDE + k] = (nn < 40) ? Wh[k * 40 + nn] : 0.0f;
  }
  __syncthreads();

  const int wave = t >> 5, lane = t & 31;
  const int row0 = (blockIdx.x * 8 + wave) * 16;
  if (row0 >= nrows) return;

  const int m = lane & 15;
  const int khalf = lane >> 4;
  v8f acc[3] = {};
  const float* Arow = H + (size_t)(row0 + m) * HID;

  for (int k0 = 0; k0 < HID; k0 += 4) {
    const int kb = k0 + 2 * khalf;
    v2f a = *(const v2f*)(Arow + kb);
#pragma unroll
    for (int nt = 0; nt < 3; ++nt) {
      v2f b = *(const v2f*)(sWT + (nt * 16 + m) * LSTRIDE + kb);
      acc[nt] = __builtin_amdgcn_wmma_f32_16x16x4_f32(
          false, a, false, b, (short)0, acc[nt], false, false);
    }
  }

#pragma unroll
  for (int nt = 0; nt < 3; ++nt) {
    const int n = nt * 16 + m;
    if (n < 40) {
      float bb = bh[n];
#pragma unroll
      for (int r = 0; r < 8; ++r)
        Out[(size_t)(row0 + khalf * 8 + r) * 40 + n] = acc[nt][r] + bb;
    }
  }
}

extern "C" void kernel_launch(void* const* d_in, const int* in_sizes, int n_in,
                              void* d_out, int out_size, void* d_ws, size_t ws_size,
                              hipStream_t stream) {
  const float* x  = (const float*)d_in[0];
  const int*   ei = (const int*)d_in[1];     // harness contract: integer -> const int*
  const float* W1 = (const float*)d_in[2];
  const float* b1 = (const float*)d_in[3];
  const float* W2 = (const float*)d_in[4];
  const float* b2 = (const float*)d_in[5];
  const float* Wh = (const float*)d_in[6];
  const float* bh = (const float*)d_in[7];

  const int n = in_sizes[0] / HID;           // 100000 (multiple of 16)
  const int e = in_sizes[1] / 2;             // 1600000
  const int* srcI = ei;                      // edge_index[0]
  const int* dstI = ei + e;                  // edge_index[1]

  float* scores = (float*)d_out;                       // [n x 40]
  float* hbuf   = (float*)d_out + (size_t)n * 40;      // [n x 128] agg/h region == final h output
  float* tbuf   = (float*)d_ws;                        // [n x 128] GEMM result (~51.2 MB)
  float* dinv   = (float*)d_ws + (size_t)n * HID;      // [n] deg, then dinv in place

  const dim3 blk(256);
  const int gemmBlocks = (n + 127) / 128;              // 8 waves x 16 rows per block
  const int aggBlocks  = (n * 32 + 255) / 256;
  const int edgeBlocks = (int)(((size_t)e * 32 + 255) / 256);
  const unsigned long long totalH = (unsigned long long)n * HID;
  const int reluBlocks = (int)((totalH + 255) / 256);

  // degrees with self-loops -> dinv = rsqrt(deg)
  k_fill1<<<(n + 255) / 256, blk, 0, stream>>>(dinv, n);
  k_deg  <<<(e + 255) / 256, blk, 0, stream>>>(dstI, dinv, e);
  k_rsqrt<<<(n + 255) / 256, blk, 0, stream>>>(dinv, n);

  // layer 1: t = x@W1 ; agg = D^-1/2 A_hat D^-1/2 t ; h1 = relu(agg + b1)
  k_gemm128 <<<gemmBlocks, blk, 0, stream>>>(x, W1, tbuf, n);
  k_agg_init<<<aggBlocks,  blk, 0, stream>>>(tbuf, dinv, hbuf, n);
  k_edge    <<<edgeBlocks, blk, 0, stream>>>(srcI, dstI, dinv, tbuf, hbuf, e);
  k_bias_relu<<<reluBlocks, blk, 0, stream>>>(hbuf, b1, totalH);

  // layer 2 (h2 lands in d_out's h region)
  k_gemm128 <<<gemmBlocks, blk, 0, stream>>>(hbuf, W2, tbuf, n);
  k_agg_init<<<aggBlocks,  blk, 0, stream>>>(tbuf, dinv, hbuf, n);
  k_edge    <<<edgeBlocks, blk, 0, stream>>>(srcI, dstI, dinv, tbuf, hbuf, e);
  k_bias_relu<<<reluBlocks, blk, 0, stream>>>(hbuf, b2, totalH);

  // classifier: scores = h2 @ Wh + bh
  k_cls<<<gemmBlocks, blk, 0, stream>>>(hbuf, Wh, bh, scores, n);
}